// MeshPool_12884901888475
// MI455X (gfx1250) — compile-verified
//
#include <hip/hip_runtime.h>

#define D_FEAT 256
#define ROWS_PER_BLOCK 8   // one wave32 per coarse row
#define CHUNK 4            // gathered x-rows staged per wave per iteration

// ---------------- CDNA5 async global->LDS helpers (inline asm) ----------------
typedef __attribute__((address_space(3))) void as3_void;

__device__ __forceinline__ void async_load_b128(const float* gsrc, float* lds_dst) {
  // LDS destination operand is the wave-relative LDS byte address in a VGPR.
  unsigned loff = (unsigned)(unsigned long long)(as3_void*)lds_dst;
  asm volatile("global_load_async_to_lds_b128 %0, %1, off"
               :: "v"(loff), "v"(gsrc)
               : "memory");
}

__device__ __forceinline__ void wait_async_zero() {
  asm volatile("s_wait_asynccnt 0" ::: "memory");
}

// ---------------- gather + weighted mean (defined first so disasm shows it) ----
// One wave32 per coarse row. Lane L owns features [4L,4L+4) and [128+4L,128+4L+4).
// Gathered x rows are staged to LDS with CDNA5 async b128 copies.
// m is forced wave-uniform via readfirstlane so all CSR bookkeeping scalarizes
// to SALU/SMEM and only the feature math stays on the VALU.
__global__ __launch_bounds__(256) void pool_gather_kernel(
    const float* __restrict__ x, const float* __restrict__ vals,
    const int* __restrict__ cols, const int* __restrict__ row_ptr,
    const int* __restrict__ perm, float* __restrict__ out, int M) {
  __shared__ float smem[ROWS_PER_BLOCK * CHUNK * D_FEAT];  // 32 KB
  const int lane = threadIdx.x & 31;
  const int m = __builtin_amdgcn_readfirstlane(
      (int)(blockIdx.x * ROWS_PER_BLOCK + (threadIdx.x >> 5)));
  if (m >= M) return;                      // wave-uniform exit
  float* buf = smem + (threadIdx.x >> 5) * (CHUNK * D_FEAT);

  const int e0  = row_ptr[m];
  const int e1  = row_ptr[m + 1];
  const int cnt = e1 - e0;

  // For small rows, sort nnz ids so accumulation order matches segment_sum
  // (deterministic regardless of atomic scatter order). Typical cnt == 4.
  int ids[8];
  const bool small = (cnt <= 8);
  if (small) {
    for (int j = 0; j < cnt; ++j) ids[j] = perm[e0 + j];
    for (int a = 1; a < cnt; ++a) {
      int key = ids[a]; int b = a - 1;
      while (b >= 0 && ids[b] > key) { ids[b + 1] = ids[b]; --b; }
      ids[b + 1] = key;
    }
  }

  float den = 0.0f;
  float4 acc0 = make_float4(0.f, 0.f, 0.f, 0.f);
  float4 acc1 = make_float4(0.f, 0.f, 0.f, 0.f);

  for (int c = 0; c < cnt; c += CHUNK) {
    const int k = (cnt - c < CHUNK) ? (cnt - c) : CHUNK;
    // Stage k gathered rows (k x 1KB) into this wave's LDS slice.
    for (int j = 0; j < k; ++j) {
      const int nz = small ? ids[c + j] : perm[e0 + c + j];
      const float* src = x + (size_t)cols[nz] * D_FEAT;
      float* dst = buf + j * D_FEAT;
      async_load_b128(src + 4 * lane,        dst + 4 * lane);
      async_load_b128(src + 128 + 4 * lane,  dst + 128 + 4 * lane);
    }
    wait_async_zero();
    // Weighted accumulation out of LDS.
    for (int j = 0; j < k; ++j) {
      const int nz = small ? ids[c + j] : perm[e0 + c + j];
      const float v = vals[nz];
      den += v;
      const float4 b0 = *(const float4*)(buf + j * D_FEAT + 4 * lane);
      const float4 b1 = *(const float4*)(buf + j * D_FEAT + 128 + 4 * lane);
      acc0.x = fmaf(v, b0.x, acc0.x); acc0.y = fmaf(v, b0.y, acc0.y);
      acc0.z = fmaf(v, b0.z, acc0.z); acc0.w = fmaf(v, b0.w, acc0.w);
      acc1.x = fmaf(v, b1.x, acc1.x); acc1.y = fmaf(v, b1.y, acc1.y);
      acc1.z = fmaf(v, b1.z, acc1.z); acc1.w = fmaf(v, b1.w, acc1.w);
    }
  }

  float4 r0, r1;
  r0.x = acc0.x / den; r0.y = acc0.y / den; r0.z = acc0.z / den; r0.w = acc0.w / den;
  r1.x = acc1.x / den; r1.y = acc1.y / den; r1.z = acc1.z / den; r1.w = acc1.w / den;
  float* orow = out + (size_t)m * D_FEAT;
  *(float4*)(orow + 4 * lane)       = r0;
  *(float4*)(orow + 128 + 4 * lane) = r1;
}

// ---------------- CSR construction ----------------
__global__ void zero_counts_kernel(int* __restrict__ counts, int M) {
  for (int i = blockIdx.x * blockDim.x + threadIdx.x; i < M; i += gridDim.x * blockDim.x)
    counts[i] = 0;
}

__global__ void count_rows_kernel(const int* __restrict__ rows, int* __restrict__ counts, int nnz) {
  for (int i = blockIdx.x * blockDim.x + threadIdx.x; i < nnz; i += gridDim.x * blockDim.x)
    atomicAdd(&counts[rows[i]], 1);
}

// Single-block exclusive scan over M counts -> row_ptr[0..M].
// Also overwrites counts[m] with row start (reused as scatter cursor).
__global__ void scan_rows_kernel(int* __restrict__ counts, int* __restrict__ row_ptr, int M) {
  __shared__ int sh[1024];
  const int tid = threadIdx.x;
  int off = 0;
  for (int base = 0; base < M; base += 1024) {
    const int idx = base + tid;
    const int v = (idx < M) ? counts[idx] : 0;
    sh[tid] = v;
    __syncthreads();
    for (int s = 1; s < 1024; s <<= 1) {
      const int t = (tid >= s) ? sh[tid - s] : 0;
      __syncthreads();
      sh[tid] += t;
      __syncthreads();
    }
    const int incl = sh[tid];
    const int tot  = sh[1023];
    if (idx < M) {
      const int e = off + incl - v;   // exclusive prefix
      row_ptr[idx] = e;
      counts[idx]  = e;               // cursor copy for scatter
    }
    off += tot;
    __syncthreads();
  }
  if (tid == 0) row_ptr[M] = off;
}

__global__ void scatter_perm_kernel(const int* __restrict__ rows, int* __restrict__ cursor,
                                    int* __restrict__ perm, int nnz) {
  for (int i = blockIdx.x * blockDim.x + threadIdx.x; i < nnz; i += gridDim.x * blockDim.x) {
    const int p = atomicAdd(&cursor[rows[i]], 1);
    perm[p] = i;
  }
}

// ---------------- launch ----------------
extern "C" void kernel_launch(void* const* d_in, const int* in_sizes, int n_in,
                              void* d_out, int out_size, void* d_ws, size_t ws_size,
                              hipStream_t stream) {
  const float* x    = (const float*)d_in[0];
  const float* vals = (const float*)d_in[1];
  const int*   rows = (const int*)d_in[2];
  const int*   cols = (const int*)d_in[3];
  float* out = (float*)d_out;

  const int nnz = in_sizes[1];
  const int M   = out_size / D_FEAT;

  // ws layout: counts[M] | row_ptr[M+1] | perm[nnz]   (~600 KB for reference sizes)
  int* counts  = (int*)d_ws;
  int* row_ptr = counts + M;
  int* perm    = row_ptr + M + 1;

  zero_counts_kernel<<<(M + 255) / 256, 256, 0, stream>>>(counts, M);
  count_rows_kernel<<<(nnz + 255) / 256, 256, 0, stream>>>(rows, counts, nnz);
  scan_rows_kernel<<<1, 1024, 0, stream>>>(counts, row_ptr, M);
  scatter_perm_kernel<<<(nnz + 255) / 256, 256, 0, stream>>>(rows, counts, perm, nnz);
  pool_gather_kernel<<<(M + ROWS_PER_BLOCK - 1) / ROWS_PER_BLOCK, 256, 0, stream>>>(
      x, vals, cols, row_ptr, perm, out, M);
}